// sub_encoder_75703093559524
// MI455X (gfx1250) — compile-verified
//
#include <hip/hip_runtime.h>

#define HDIM 128
#define VOCABSZ 800

typedef __attribute__((ext_vector_type(2))) float v2f;
typedef __attribute__((ext_vector_type(8))) float v8f;

// ---------------------------------------------------------------------------
// Zero a float buffer (grid-stride).
// ---------------------------------------------------------------------------
__global__ void zero_f32(float* __restrict__ p, long long n) {
  long long i = (long long)blockIdx.x * blockDim.x + threadIdx.x;
  long long stride = (long long)gridDim.x * blockDim.x;
  for (; i < n; i += stride) p[i] = 0.0f;
}

// ---------------------------------------------------------------------------
// Kernel 1: per-edge scatter-add of x[row[e]] into frag_sum[col[e]].
// One wave (32 lanes) per edge; each lane moves 4 contiguous floats.
// Also accumulates per-fragment and per-node edge counts (lane 0).
// Random-index atomics resolve in the 192MB L2 (whole working set resident).
// ---------------------------------------------------------------------------
__global__ void edge_scatter_to_frag(const float* __restrict__ x,
                                     const int* __restrict__ row,
                                     const int* __restrict__ col,
                                     float* __restrict__ frag_sum,
                                     float* __restrict__ frag_cnt,
                                     float* __restrict__ node_cnt,
                                     int n_edges) {
  int wave = threadIdx.x >> 5;
  int lane = threadIdx.x & 31;
  int e = blockIdx.x * 8 + wave;
  if (e >= n_edges) return;
  int r = row[e];
  int c = col[e];
  const float4 v = *(const float4*)(x + (long long)r * HDIM + lane * 4);
  float* dst = frag_sum + (long long)c * HDIM + lane * 4;
  atomicAdd(dst + 0, v.x);
  atomicAdd(dst + 1, v.y);
  atomicAdd(dst + 2, v.z);
  atomicAdd(dst + 3, v.w);
  if (lane == 0) {
    atomicAdd(frag_cnt + c, 1.0f);
    atomicAdd(node_cnt + r, 1.0f);
  }
}

// ---------------------------------------------------------------------------
// Kernel 2: wave-per-fragment argmax over VOCAB=800 (streaming, BW-bound).
// Shuffle tree reduction, first-max tie-break to match jnp.argmax.
// ---------------------------------------------------------------------------
__global__ void frag_argmax(const float* __restrict__ fragments,
                            int* __restrict__ frag_ids, int n_frags) {
  int wave = threadIdx.x >> 5;
  int lane = threadIdx.x & 31;
  int f = blockIdx.x * 8 + wave;
  if (f >= n_frags) return;
  const float* rowp = fragments + (long long)f * VOCABSZ;
  float best = -3.402823466e38f;
  int bi = 0x7fffffff;
  for (int j = lane; j < VOCABSZ; j += 32) {
    float v = rowp[j];
    if (v > best || (v == best && j < bi)) { best = v; bi = j; }
  }
  for (int off = 16; off >= 1; off >>= 1) {
    float ov = __shfl_xor(best, off, 32);
    int oi = __shfl_xor(bi, off, 32);
    if (ov > best || (ov == best && oi < bi)) { best = ov; bi = oi; }
  }
  if (lane == 0) frag_ids[f] = bi;
}

// ---------------------------------------------------------------------------
// One K=4 f32 WMMA step, operands loaded per the CDNA5 16x16x4 F32 layouts:
//   A 16x4 : lanes 0-15 (m=lane) hold K={k,k+1}; lanes 16-31 hold K={k+2,k+3}
//   B 4x16 : lanes 0-15 (n=lane) hold K={k,k+1}; lanes 16-31 hold K={k+2,k+3}
// ---------------------------------------------------------------------------
__device__ inline v8f wmma_step_f32(const float* arow, float ascale,
                                    const float* bmat, int ldb, int kk,
                                    int kh, int ncol, v8f c) {
  v2f a;
  a.x = arow[kk + 2 * kh] * ascale;
  a.y = arow[kk + 2 * kh + 1] * ascale;
  const float* bp = bmat + (long long)(kk + 2 * kh) * ldb + ncol;
  v2f b;
  b.x = bp[0];
  b.y = bp[ldb];
  return __builtin_amdgcn_wmma_f32_16x16x4_f32(false, a, false, b, (short)0, c,
                                               false, false);
}

// ---------------------------------------------------------------------------
// Kernel 3: fold Wc = W_f @ W_s[128:256,:]  (128x128 @ 128x128, exact f32 WMMA).
// 64 waves total; each wave computes one 16x16 tile, K=128 (32 wmma ops).
// ---------------------------------------------------------------------------
__global__ void fold_weight_gemm(const float* __restrict__ W_f,
                                 const float* __restrict__ W_s,
                                 float* __restrict__ Wc) {
  const float* Wsb = W_s + HDIM * HDIM;  // bottom half of [256,128]
  int lane = threadIdx.x & 31;
  int wave = threadIdx.x >> 5;
  int tile = blockIdx.x * 8 + wave;      // 0..63
  int mt = tile >> 3;
  int nt = tile & 7;
  int m = lane & 15;
  int kh = lane >> 4;
  int ncol = nt * 16 + (lane & 15);
  const float* arow = W_f + (long long)(mt * 16 + m) * HDIM;
  v8f c = {};
  for (int kk = 0; kk < HDIM; kk += 4)
    c = wmma_step_f32(arow, 1.0f, Wsb, HDIM, kk, kh, ncol, c);
  for (int r2 = 0; r2 < 8; ++r2)
    Wc[(long long)(mt * 16 + r2 + 8 * kh) * HDIM + ncol] = c[r2];
}

// ---------------------------------------------------------------------------
// Kernel 4: fold bc = b_f @ W_s[128:256,:] + b_s   (128 outputs, trivial).
// ---------------------------------------------------------------------------
__global__ void fold_bias(const float* __restrict__ b_f,
                          const float* __restrict__ W_s,
                          const float* __restrict__ b_s,
                          float* __restrict__ bc) {
  int n = threadIdx.x;  // 128 threads
  const float* Wsb = W_s + HDIM * HDIM;
  float acc = b_s[n];
  for (int k = 0; k < HDIM; ++k) acc = fmaf(b_f[k], Wsb[k * HDIM + n], acc);
  bc[n] = acc;
}

// ---------------------------------------------------------------------------
// Kernel 5: fused per-fragment GEMM (exact f32 via V_WMMA_F32_16X16X4_F32):
//   frag_emb = (frag_sum/cnt) @ Ws_top + emb_table[frag_id] @ Wc + bc
// Block = 16 fragments x 128 cols; 8 waves, each wave a 16x16 tile,
// 64 wmma ops per wave (two K=128 passes). Mean-scale folded into A load.
// ---------------------------------------------------------------------------
__global__ void frag_gemm(const float* __restrict__ frag_sum,
                          const float* __restrict__ frag_cnt,
                          const int* __restrict__ frag_ids,
                          const float* __restrict__ emb_table,
                          const float* __restrict__ W_s,   // top half used
                          const float* __restrict__ Wc,
                          const float* __restrict__ bc,
                          float* __restrict__ frag_emb) {
  int lane = threadIdx.x & 31;
  int wave = threadIdx.x >> 5;   // 8 waves -> 8 N-tiles of 16
  int f0 = blockIdx.x * 16;
  int m = lane & 15;
  int kh = lane >> 4;
  int ncol = wave * 16 + (lane & 15);
  int fm = f0 + m;

  float rinv = 1.0f / fmaxf(frag_cnt[fm], 1.0f);
  const float* arow = frag_sum + (long long)fm * HDIM;
  const float* erow = emb_table + (long long)frag_ids[fm] * HDIM;

  v8f c = {};
  for (int kk = 0; kk < HDIM; kk += 4)            // S @ Ws_top (mean folded)
    c = wmma_step_f32(arow, rinv, W_s, HDIM, kk, kh, ncol, c);
  for (int kk = 0; kk < HDIM; kk += 4)            // E_gathered @ Wc
    c = wmma_step_f32(erow, 1.0f, Wc, HDIM, kk, kh, ncol, c);

  float bias = bc[ncol];
  for (int r2 = 0; r2 < 8; ++r2)
    frag_emb[(long long)(f0 + r2 + 8 * kh) * HDIM + ncol] = c[r2] + bias;
}

// ---------------------------------------------------------------------------
// Kernel 6: per-edge scatter-add of frag_emb[col[e]] into out[row[e]].
// ---------------------------------------------------------------------------
__global__ void edge_scatter_to_node(const float* __restrict__ frag_emb,
                                     const int* __restrict__ row,
                                     const int* __restrict__ col,
                                     float* __restrict__ out, int n_edges) {
  int wave = threadIdx.x >> 5;
  int lane = threadIdx.x & 31;
  int e = blockIdx.x * 8 + wave;
  if (e >= n_edges) return;
  int r = row[e];
  int c = col[e];
  const float4 v = *(const float4*)(frag_emb + (long long)c * HDIM + lane * 4);
  float* dst = out + (long long)r * HDIM + lane * 4;
  atomicAdd(dst + 0, v.x);
  atomicAdd(dst + 1, v.y);
  atomicAdd(dst + 2, v.z);
  atomicAdd(dst + 3, v.w);
}

// ---------------------------------------------------------------------------
// Kernel 7: finalize node means: out /= max(node_cnt, 1).
// ---------------------------------------------------------------------------
__global__ void node_mean(float* __restrict__ out,
                          const float* __restrict__ node_cnt, long long n) {
  long long i = (long long)blockIdx.x * blockDim.x + threadIdx.x;
  if (i >= n) return;
  float cnt = node_cnt[i >> 7];  // HDIM == 128
  out[i] *= 1.0f / fmaxf(cnt, 1.0f);
}

// ---------------------------------------------------------------------------
extern "C" void kernel_launch(void* const* d_in, const int* in_sizes, int n_in,
                              void* d_out, int out_size, void* d_ws,
                              size_t ws_size, hipStream_t stream) {
  const float* x         = (const float*)d_in[0];
  const float* fragments = (const float*)d_in[1];
  const float* emb_table = (const float*)d_in[2];
  const float* W_f       = (const float*)d_in[3];
  const float* b_f       = (const float*)d_in[4];
  const float* W_s       = (const float*)d_in[5];
  const float* b_s       = (const float*)d_in[6];
  const int*   row       = (const int*)d_in[7];
  const int*   col       = (const int*)d_in[8];
  float* out = (float*)d_out;

  const long long N = in_sizes[0] / HDIM;     // 200000
  const long long F = in_sizes[1] / VOCABSZ;  // 50000
  const int       E = in_sizes[7];            // 400000

  // Workspace layout (floats), zero-init region first and contiguous:
  //   [frag_sum F*128 | frag_cnt F | node_cnt N] | frag_emb F*128 |
  //   Wc 128*128 | bc 128 | frag_ids (int) F
  float* ws        = (float*)d_ws;
  float* frag_sum  = ws;
  float* frag_cnt  = frag_sum + F * HDIM;
  float* node_cnt  = frag_cnt + F;
  float* frag_emb  = node_cnt + N;
  float* Wc        = frag_emb + F * HDIM;
  float* bcv       = Wc + HDIM * HDIM;
  int*   frag_ids  = (int*)(bcv + HDIM);

  const long long zero_n = F * HDIM + F + N;
  zero_f32<<<4096, 256, 0, stream>>>(frag_sum, zero_n);
  zero_f32<<<4096, 256, 0, stream>>>(out, (long long)N * HDIM);

  // 1) scatter x[row] -> frag sums (+counts)
  edge_scatter_to_frag<<<(E + 7) / 8, 256, 0, stream>>>(
      x, row, col, frag_sum, frag_cnt, node_cnt, E);

  // 2) fragment-type argmax
  frag_argmax<<<(int)((F + 7) / 8), 256, 0, stream>>>(fragments, frag_ids,
                                                      (int)F);

  // 3) fold weights/bias:  Wc = W_f @ Ws_bot ;  bc = b_f @ Ws_bot + b_s
  fold_weight_gemm<<<8, 256, 0, stream>>>(W_f, W_s, Wc);
  fold_bias<<<1, HDIM, 0, stream>>>(b_f, W_s, b_s, bcv);

  // 4) fused fragment GEMM (F is a multiple of 16: 50000/16 = 3125 blocks)
  frag_gemm<<<(int)(F / 16), 256, 0, stream>>>(frag_sum, frag_cnt, frag_ids,
                                               emb_table, W_s, Wc, bcv,
                                               frag_emb);

  // 5) scatter frag_emb[col] -> node sums
  edge_scatter_to_node<<<(E + 7) / 8, 256, 0, stream>>>(frag_emb, row, col,
                                                        out, E);

  // 6) divide by per-node counts
  const long long outn = N * HDIM;
  node_mean<<<(int)((outn + 255) / 256), 256, 0, stream>>>(out, node_cnt,
                                                           outn);
}